// GATLayer_16123307229362
// MI455X (gfx1250) — compile-verified
//
#include <hip/hip_runtime.h>
#include <hip/hip_bf16.h>

#define ALPHA   0.2f
#define NEG_INF (-9000000000000000.0f)

// Problem constants (B=4, T=32, N=512, F=128, D=128)
#define BT 128
#define NN 512
#define FF 128
#define DD 128
#define LDSP  132  // padded stride (floats) for row-major sH: bank = (4*row+col)%64
#define PAIRP 144  // padded stride (float2) for K-pair layouts: 288 dwords == 32 mod 64 banks

typedef float v2f __attribute__((ext_vector_type(2)));
typedef float v8f __attribute__((ext_vector_type(8)));

// ---------------------------------------------------------------------------
// K1: Wh = h @ W   (per (bt, 128-row block); WMMA f32 16x16x4)
// grid = BT*4 blocks, 256 threads (8 waves)
// dyn LDS: sH [128][LDSP] floats + sW pairs [64][PAIRP] float2
// ---------------------------------------------------------------------------
__global__ void gat_wh_kernel(const float* __restrict__ h,
                              const float* __restrict__ W,
                              float* __restrict__ Wh) {
  extern __shared__ float smem[];
  float* sH = smem;                      // [128][LDSP]
  v2f*   sW = (v2f*)(smem + 128 * LDSP); // [64][PAIRP] : {W[k][n], W[k+1][n]}

  const int bt = blockIdx.x >> 2;
  const int rb = blockIdx.x & 3;
  const float* hblk  = h  + ((size_t)bt * NN + (size_t)rb * 128) * FF;
  float*       whblk = Wh + ((size_t)bt * NN + (size_t)rb * 128) * DD;

  const int tid = threadIdx.x;
  // stage h tile row-major (contiguous global chunk -> padded LDS rows)
  for (int idx = tid * 4; idx < 128 * FF; idx += 256 * 4) {
    const int r = idx >> 7, c = idx & 127;
    *(float4*)(sH + r * LDSP + c) = *(const float4*)(hblk + idx);
  }
  // stage W in K-pair-interleaved layout: 64 pair-rows x 32 n-quads
  for (int idx = tid; idx < 64 * 32; idx += 256) {
    const int k2 = idx >> 5, nq = idx & 31;
    float4 r0 = *(const float4*)(W + (2 * k2) * DD + nq * 4);
    float4 r1 = *(const float4*)(W + (2 * k2 + 1) * DD + nq * 4);
    float* dst = (float*)(sW + k2 * PAIRP + nq * 4);
    *(float4*)(dst)     = make_float4(r0.x, r1.x, r0.y, r1.y);
    *(float4*)(dst + 4) = make_float4(r0.z, r1.z, r0.w, r1.w);
  }
  __syncthreads();

  const int lane = tid & 31;
  const int wave = tid >> 5;
  const int l16  = lane & 15;
  const int g    = lane >> 4;
  const int arow = wave * 16 + l16;

  v8f acc[8];
#pragma unroll
  for (int t = 0; t < 8; ++t)
#pragma unroll
    for (int r = 0; r < 8; ++r) acc[t][r] = 0.0f;

  for (int kk = 0; kk < FF; kk += 4) {
    const int ka = kk + 2 * g;  // lane's K pair: ka, ka+1 (contiguous in sH row)
    const v2f a = *(const v2f*)(sH + arow * LDSP + ka);
    const v2f* brow = sW + ((kk >> 1) + g) * PAIRP + l16;
    // phase 1: batch all B-fragment loads (4x ds_load_2addr_b64 in flight)
    v2f b[8];
#pragma unroll
    for (int t = 0; t < 8; ++t) b[t] = brow[t * 16];
    // phase 2: matrix pipe
#pragma unroll
    for (int t = 0; t < 8; ++t)
      acc[t] = __builtin_amdgcn_wmma_f32_16x16x4_f32(
          false, a, false, b[t], (short)0, acc[t], false, false);
  }

#pragma unroll
  for (int t = 0; t < 8; ++t)
#pragma unroll
    for (int r = 0; r < 8; ++r)
      whblk[(wave * 16 + r + 8 * g) * DD + t * 16 + l16] = acc[t][r];
}

// ---------------------------------------------------------------------------
// K2: e_src / e_dst row dot products; one wave per row
// ---------------------------------------------------------------------------
__global__ void gat_e_kernel(const float* __restrict__ Wh,
                             const float* __restrict__ a,
                             float* __restrict__ esrc,
                             float* __restrict__ edst) {
  const int lane = threadIdx.x & 31;
  const int row  = (blockIdx.x * blockDim.x + threadIdx.x) >> 5;

  float4 v  = *(const float4*)(Wh + (size_t)row * DD + lane * 4);
  float4 as = *(const float4*)(a + lane * 4);
  float4 ad = *(const float4*)(a + DD + lane * 4);
  float ps = v.x * as.x + v.y * as.y + v.z * as.z + v.w * as.w;
  float pd = v.x * ad.x + v.y * ad.y + v.z * ad.z + v.w * ad.w;
#pragma unroll
  for (int off = 16; off >= 1; off >>= 1) {
    ps += __shfl_xor(ps, off, 32);
    pd += __shfl_xor(pd, off, 32);
  }
  if (lane == 0) {
    esrc[row] = ps;
    edst[row] = pd;
  }
}

// ---------------------------------------------------------------------------
// K3: fused masked-softmax + attn@Wh
// grid = BT*4 blocks (128 out rows), 256 threads (8 waves, 16 rows/wave)
// dyn LDS: Wh pairs [256][PAIRP] float2 + e_dst[512] + masks 8*16*16 words
// ---------------------------------------------------------------------------
__global__ void gat_attn_kernel(const float* __restrict__ Wh,
                                const float* __restrict__ esrc,
                                const float* __restrict__ edst,
                                const int* __restrict__ adj,
                                float* __restrict__ out) {
  extern __shared__ float smem[];
  v2f*      sWh   = (v2f*)smem;                       // [256][PAIRP] pairs
  float*    sEd   = smem + (NN / 2) * PAIRP * 2;      // [512]
  unsigned* sMask = (unsigned*)(sEd + NN);            // [8 waves][16 rows][16 words]

  const int bt = blockIdx.x >> 2;
  const int rb = blockIdx.x & 3;
  const int tid = threadIdx.x;
  const float* whbt = Wh + (size_t)bt * NN * DD;

  // stage Wh slice in K-pair-interleaved layout: 256 pair-rows x 32 n-quads
  for (int idx = tid; idx < 256 * 32; idx += 256) {
    const int k2 = idx >> 5, nq = idx & 31;
    float4 r0 = *(const float4*)(whbt + (size_t)(2 * k2) * DD + nq * 4);
    float4 r1 = *(const float4*)(whbt + (size_t)(2 * k2 + 1) * DD + nq * 4);
    float* dst = (float*)(sWh + k2 * PAIRP + nq * 4);
    *(float4*)(dst)     = make_float4(r0.x, r1.x, r0.y, r1.y);
    *(float4*)(dst + 4) = make_float4(r0.z, r1.z, r0.w, r1.w);
  }
  for (int j = tid; j < NN; j += 256) sEd[j] = edst[bt * NN + j];
  __syncthreads();

  const int lane = tid & 31;
  const int wave = tid >> 5;
  const int l16  = lane & 15;
  const int g    = lane >> 4;

  const int irow = rb * 128 + wave * 16 + l16;  // this lane's row
  const float es = esrc[bt * NN + irow];
  const int* adjrow = adj + irow * NN;
  unsigned* maskrow = sMask + (wave * 16 + l16) * 16;

  // ---- pass 1: online max/sum + build 512-bit adjacency mask per row ----
  float m = -3.0e38f, l = 0.0f;
  for (int w = 0; w < 16; ++w) {
    unsigned mw = 0;
#pragma unroll
    for (int jj = 0; jj < 16; ++jj) {
      const int j = w * 32 + 2 * jj + g;
      const bool conn = adjrow[j] > 0;
      float ev = es + sEd[j];
      ev = ev > 0.0f ? ev : ALPHA * ev;
      if (!conn) ev = NEG_INF;
      mw |= (conn ? 1u : 0u) << (2 * jj + g);
      const float nm = fmaxf(m, ev);
      l = l * __expf(m - nm) + __expf(ev - nm);
      m = nm;
    }
    mw |= __shfl_xor(mw, 16, 32);
    if (g == 0) maskrow[w] = mw;
  }
  {
    const float m2 = __shfl_xor(m, 16, 32);
    const float l2 = __shfl_xor(l, 16, 32);
    const float nm = fmaxf(m, m2);
    l = l * __expf(m - nm) + l2 * __expf(m2 - nm);
    m = nm;
  }

  // ---- pass 2: acc(16x128) += P(16x4) @ Wh(4x128), mask from LDS bits ----
  v8f acc[8];
#pragma unroll
  for (int t = 0; t < 8; ++t)
#pragma unroll
    for (int r = 0; r < 8; ++r) acc[t][r] = 0.0f;

  for (int w = 0; w < 16; ++w) {
    const unsigned mw = maskrow[w];
#pragma unroll
    for (int c = 0; c < 8; ++c) {
      const int j0 = w * 32 + c * 4;
      const int ja = j0 + 2 * g;
      float e0 = es + sEd[ja];
      e0 = e0 > 0.0f ? e0 : ALPHA * e0;
      if (!((mw >> (c * 4 + 2 * g)) & 1u)) e0 = NEG_INF;
      float e1 = es + sEd[ja + 1];
      e1 = e1 > 0.0f ? e1 : ALPHA * e1;
      if (!((mw >> (c * 4 + 2 * g + 1)) & 1u)) e1 = NEG_INF;
      v2f a;
      a.x = __expf(e0 - m);
      a.y = __expf(e1 - m);
      const v2f* brow = sWh + ((j0 >> 1) + g) * PAIRP + l16;
      // phase 1: batch B-fragment loads; phase 2: WMMA
      v2f b[8];
#pragma unroll
      for (int t = 0; t < 8; ++t) b[t] = brow[t * 16];
#pragma unroll
      for (int t = 0; t < 8; ++t)
        acc[t] = __builtin_amdgcn_wmma_f32_16x16x4_f32(
            false, a, false, b[t], (short)0, acc[t], false, false);
    }
  }

  // ---- epilogue: scale by 1/l, store ----
  float* outblk = out + ((size_t)bt * NN + (size_t)rb * 128 + wave * 16) * DD;
#pragma unroll
  for (int r = 0; r < 8; ++r) {
    const int orow = r + 8 * g;
    const float lr = __shfl(l, orow, 32);
    const float inv = 1.0f / lr;
#pragma unroll
    for (int t = 0; t < 8; ++t)
      outblk[orow * DD + t * 16 + l16] = acc[t][r] * inv;
  }
}

// ---------------------------------------------------------------------------
extern "C" void kernel_launch(void* const* d_in, const int* in_sizes, int n_in,
                              void* d_out, int out_size, void* d_ws, size_t ws_size,
                              hipStream_t stream) {
  const float* h   = (const float*)d_in[0];  // [128][512][128]
  const int*   adj = (const int*)d_in[1];    // [512][512]
  const float* W   = (const float*)d_in[2];  // [128][128]
  const float* a   = (const float*)d_in[3];  // [256]
  float* out = (float*)d_out;                // [128][512][128]

  float* Wh   = (float*)d_ws;                // 8,388,608 floats (33.5 MB)
  float* esrc = Wh + (size_t)BT * NN * DD;   // 65,536 floats
  float* edst = esrc + (size_t)BT * NN;      // 65,536 floats

  const size_t lds1 = (size_t)(128 * LDSP) * 4 + (size_t)(64 * PAIRP) * 8;
  const size_t lds3 = (size_t)(NN / 2) * PAIRP * 8 + (size_t)NN * 4 +
                      (size_t)(8 * 16 * 16) * 4;

  gat_wh_kernel<<<dim3(BT * 4), dim3(256), lds1, stream>>>(h, W, Wh);
  gat_e_kernel<<<dim3(BT * NN / 8), dim3(256), 0, stream>>>(Wh, a, esrc, edst);
  gat_attn_kernel<<<dim3(BT * 4), dim3(256), lds3, stream>>>(Wh, esrc, edst, adj, out);
}